// ScaledDotProductAttention_88888643158481
// MI455X (gfx1250) — compile-verified
//
#include <hip/hip_runtime.h>

typedef __attribute__((ext_vector_type(2))) float v2f;
typedef __attribute__((ext_vector_type(4))) float v4f;
typedef __attribute__((ext_vector_type(8))) float v8f;
typedef int v4i __attribute__((vector_size(4 * sizeof(int))));

#define B_    2
#define H_    16
#define N_    2048
#define D_    64
#define BH_   (B_ * H_)
#define SCALE 0.125f   // 1/sqrt(64)

// ---------------------------------------------------------------------------
// Async global->LDS staging (CDNA5 GLOBAL_LOAD_ASYNC_TO_LDS_B128, ASYNCcnt),
// with a synchronous fallback if the builtin is not declared by this toolchain.
// Probe round 2 revealed the builtin's param 0 type:
//   'int __attribute__((vector_size(16))) __device__ *'  (AS1, non-const)
// ---------------------------------------------------------------------------
#if defined(__gfx1250__) && __has_builtin(__builtin_amdgcn_global_load_async_to_lds_b128)
#define HAVE_ASYNC_LDS 1
#else
#define HAVE_ASYNC_LDS 0
#endif

#if HAVE_ASYNC_LDS
typedef __attribute__((address_space(1))) v4i gv4i_t;   // global 16B chunk
typedef __attribute__((address_space(3))) v4i lv4i_t;   // LDS   16B chunk
#endif

__device__ __forceinline__ void copy16B_to_lds(const float* __restrict__ gp,
                                               float* __restrict__ lp) {
#if HAVE_ASYNC_LDS
    __builtin_amdgcn_global_load_async_to_lds_b128(
        (gv4i_t*)gp,
        (lv4i_t*)lp,
        0, 0);
#else
    *(v4f*)lp = *(const v4f*)gp;
#endif
}

__device__ __forceinline__ void wait_lds_copies() {
#if HAVE_ASYNC_LDS
#if __has_builtin(__builtin_amdgcn_s_wait_asynccnt)
    __builtin_amdgcn_s_wait_asynccnt(0);
#else
    asm volatile("s_wait_asynccnt 0x0" ::: "memory");
#endif
#endif
    __syncthreads();
}

// ---------------------------------------------------------------------------
// Kernel 1: raw scores  S = (SCALE*Q) K^T   -> attn buffer (bias added in k2)
// grid (N/128 cols, N/128 rows, BH), block 256 (8 waves)
// ---------------------------------------------------------------------------
#define KSTR1 68   // (c*68+d)*4: 272B row pitch -> 16B aligned chunks; banks 4c+k

__global__ __launch_bounds__(256) void scores_kernel(
    const float* __restrict__ Q, const float* __restrict__ K,
    float* __restrict__ attn)
{
    __shared__ float lK[128 * KSTR1];

    const int bh   = blockIdx.z;
    const int row0 = blockIdx.y * 128;
    const int col0 = blockIdx.x * 128;
    const int t    = threadIdx.x;

    // stage K tile: 128 cols x 64 d (coalesced 16B chunks, async if available)
    const float* Kbase = K + (size_t)bh * N_ * D_ + (size_t)col0 * D_;
    #pragma unroll
    for (int it = 0; it < 8; ++it) {
        int f = (it * 256 + t) * 4;
        int c = f >> 6, d = f & 63;
        copy16B_to_lds(Kbase + c * D_ + d, &lK[c * KSTR1 + d]);
    }
    wait_lds_copies();

    const int wave  = t >> 5;
    const int lane  = t & 31;
    const int mrow  = lane & 15;
    const int khalf = (lane >> 4) * 2;

    // A fragments (scale folded in): 16 k-steps of float2
    const int row = row0 + wave * 16 + mrow;
    const float* Qrow = Q + (size_t)bh * N_ * D_ + (size_t)row * D_;
    v2f aq[16];
    #pragma unroll
    for (int ks = 0; ks < 16; ++ks) {
        v2f q = *(const v2f*)(Qrow + ks * 4 + khalf);
        aq[ks] = q * SCALE;
    }

    const int mbase = row0 + wave * 16 + ((lane >> 4) << 3);

    for (int jt = 0; jt < 8; ++jt) {
        const float* bcol = &lK[(jt * 16 + mrow) * KSTR1];
        v8f acc = {};
        #pragma unroll
        for (int ks = 0; ks < 16; ++ks) {
            v2f b = *(const v2f*)(bcol + ks * 4 + khalf);
            acc = __builtin_amdgcn_wmma_f32_16x16x4_f32(
                      false, aq[ks], false, b, (short)0, acc, false, false);
        }
        const int ncol = col0 + jt * 16 + mrow;
        float* optr = attn + (size_t)bh * N_ * N_ + (size_t)mbase * N_ + ncol;
        #pragma unroll
        for (int r = 0; r < 8; ++r)
            optr[(size_t)r * N_] = acc[r];
    }
}

// ---------------------------------------------------------------------------
// Kernel 2: attn = softmax(raw + bias) in place; one block per row
// bias row read is contiguous here (coalesced), unlike in the GEMM epilogue.
// ---------------------------------------------------------------------------
__global__ __launch_bounds__(256) void softmax_kernel(
    float* __restrict__ attn, const float* __restrict__ bias)
{
    __shared__ float red[256];
    const int rowid = blockIdx.x;            // bh*N + m
    const int h = (rowid >> 11) & (H_ - 1);
    const int m = rowid & (N_ - 1);
    float* rowp = attn + (size_t)rowid * N_;
    const float* bp = bias + ((size_t)h * N_ + m) * N_;
    const int t = threadIdx.x;

    v4f x0 = *(const v4f*)(rowp + t * 8);
    v4f x1 = *(const v4f*)(rowp + t * 8 + 4);
    v4f b0 = *(const v4f*)(bp + t * 8);
    v4f b1 = *(const v4f*)(bp + t * 8 + 4);
    x0 += b0;
    x1 += b1;

    float mx = x0.x;
    mx = fmaxf(mx, x0.y); mx = fmaxf(mx, x0.z); mx = fmaxf(mx, x0.w);
    mx = fmaxf(mx, x1.x); mx = fmaxf(mx, x1.y); mx = fmaxf(mx, x1.z); mx = fmaxf(mx, x1.w);
    red[t] = mx;
    __syncthreads();
    #pragma unroll
    for (int s = 128; s > 0; s >>= 1) {
        if (t < s) red[t] = fmaxf(red[t], red[t + s]);
        __syncthreads();
    }
    mx = red[0];
    __syncthreads();

    x0.x = __expf(x0.x - mx); x0.y = __expf(x0.y - mx);
    x0.z = __expf(x0.z - mx); x0.w = __expf(x0.w - mx);
    x1.x = __expf(x1.x - mx); x1.y = __expf(x1.y - mx);
    x1.z = __expf(x1.z - mx); x1.w = __expf(x1.w - mx);

    red[t] = x0.x + x0.y + x0.z + x0.w + x1.x + x1.y + x1.z + x1.w;
    __syncthreads();
    #pragma unroll
    for (int s = 128; s > 0; s >>= 1) {
        if (t < s) red[t] += red[t + s];
        __syncthreads();
    }
    const float inv = 1.0f / red[0];

    x0 *= inv;
    x1 *= inv;
    *(v4f*)(rowp + t * 8)     = x0;
    *(v4f*)(rowp + t * 8 + 4) = x1;
}

// ---------------------------------------------------------------------------
// Kernel 3: O = P V. grid (N/128, BH), block 256 (8 waves).
// P (128x64) and V (64x64) k-tiles double-buffered in LDS via async copies.
// Stride 72: 16B-aligned chunks; bank = (8*row + k) mod 64 -> both lane halves
// land in disjoint bank sets for A b64 reads and B b32 pair reads.
// ---------------------------------------------------------------------------
#define KS3 72
#define KB3 64   // k-depth per pipeline stage

__global__ __launch_bounds__(256) void pv_kernel(
    const float* __restrict__ attn, const float* __restrict__ V,
    float* __restrict__ O)
{
    __shared__ float lP[2 * 128 * KS3];   // [buf][row][k]
    __shared__ float lV[2 * 64 * KS3];    // [buf][k][d]

    const int bh   = blockIdx.y;
    const int row0 = blockIdx.x * 128;
    const int t    = threadIdx.x;
    const int wave = t >> 5, lane = t & 31;
    const int mrow = lane & 15;
    const int khalf = (lane >> 4) * 2;

    const float* Pg = attn + (size_t)bh * N_ * N_;
    const float* Vg = V    + (size_t)bh * N_ * D_;

    // --- stage issue helpers (256 threads, 16B chunks, coalesced) ---
    auto issueP = [&](int buf, int kb) {
        float* dst = &lP[buf * 128 * KS3];
        #pragma unroll
        for (int it = 0; it < 8; ++it) {            // 128 rows x 64 k
            int f = (it * 256 + t) * 4;
            int r = f >> 6, k = f & 63;
            copy16B_to_lds(Pg + (size_t)(row0 + r) * N_ + kb + k, dst + r * KS3 + k);
        }
    };
    auto issueV = [&](int buf, int kb) {
        float* dst = &lV[buf * 64 * KS3];
        #pragma unroll
        for (int it = 0; it < 4; ++it) {            // 64 k x 64 d
            int f = (it * 256 + t) * 4;
            int kk = f >> 6, d = f & 63;
            copy16B_to_lds(Vg + (size_t)(kb + kk) * D_ + d, dst + kk * KS3 + d);
        }
    };

    issueP(0, 0);
    issueV(0, 0);

    v8f acc[4] = {};

    for (int i = 0; i < N_ / KB3; ++i) {
        wait_lds_copies();                 // own asyncs done, then barrier
        if (i + 1 < N_ / KB3) {            // prefetch next stage into other buffer
            issueP((i + 1) & 1, (i + 1) * KB3);
            issueV((i + 1) & 1, (i + 1) * KB3);
        }
        const float* Pb = &lP[(i & 1) * 128 * KS3];
        const float* Vb = &lV[(i & 1) * 64 * KS3];
        const float* arow = Pb + (wave * 16 + mrow) * KS3;

        #pragma unroll
        for (int ks = 0; ks < 16; ++ks) {
            v2f a = *(const v2f*)(arow + ks * 4 + khalf);
            const int k0 = ks * 4 + khalf;
            const float* vb = Vb + k0 * KS3 + mrow;   // pair at +0, +KS3 -> ds_load_2addr
            #pragma unroll
            for (int jt = 0; jt < 4; ++jt) {
                v2f b;
                b.x = vb[jt * 16];
                b.y = vb[jt * 16 + KS3];
                acc[jt] = __builtin_amdgcn_wmma_f32_16x16x4_f32(
                              false, a, false, b, (short)0, acc[jt], false, false);
            }
        }
    }

    const int mbase = row0 + wave * 16 + ((lane >> 4) << 3);
    float* Ob = O + (size_t)bh * N_ * D_ + (size_t)mbase * D_;
    #pragma unroll
    for (int jt = 0; jt < 4; ++jt) {
        const int ncol = jt * 16 + mrow;
        #pragma unroll
        for (int r = 0; r < 8; ++r)
            Ob[(size_t)r * D_ + ncol] = acc[jt][r];
    }
}

// ---------------------------------------------------------------------------
extern "C" void kernel_launch(void* const* d_in, const int* in_sizes, int n_in,
                              void* d_out, int out_size, void* d_ws, size_t ws_size,
                              hipStream_t stream) {
    (void)in_sizes; (void)n_in; (void)out_size; (void)d_ws; (void)ws_size;
    const float* Q    = (const float*)d_in[0];
    const float* K    = (const float*)d_in[1];
    const float* V    = (const float*)d_in[2];
    const float* bias = (const float*)d_in[3];

    float* out  = (float*)d_out;                        // (B,H,N,D)
    float* attn = out + (size_t)B_ * H_ * N_ * D_;      // (B,H,N,N)

    dim3 g1(N_ / 128, N_ / 128, BH_);
    scores_kernel<<<g1, 256, 0, stream>>>(Q, K, attn);

    softmax_kernel<<<dim3(BH_ * N_), 256, 0, stream>>>(attn, bias);

    pv_kernel<<<dim3(N_ / 128, BH_), 256, 0, stream>>>(attn, V, out);
}